// IF_23304492548821
// MI455X (gfx1250) — compile-verified
//
#include <hip/hip_runtime.h>

// IF spiking neuron forward pass for MI455X (gfx1250).
//
// Reference forward: temporal scan over T=8 of
//   mem += x[t]; out[t] = (mem > 1) ? 1 : 0; mem = out[t] ? 0 : mem;
// (STE surrogate term is exactly out_s in fp32 forward arithmetic:
//  Sterbenz => (1-b)+b == 1 and (0-b)+b == 0 for b in the clip range.)
//
// Pure streaming workload: 512 MB HBM traffic, ~22 us floor at 23.3 TB/s.
// No matmul structure -> WMMA inapplicable. Levers:
//   - B128 (float4) loads/stores, non-temporal policy (zero reuse, 512 MB > L2)
//   - 8-deep per-wave MLP: all timestep loads issued before the sequential
//     compute chain, enforced with sched_barrier so the scheduler cannot
//     sink loads behind dependent compute (round-1 asm showed a full
//     s_wait_loadcnt 0x0 stall after the first load otherwise).

typedef float v4f __attribute__((ext_vector_type(4)));

constexpr int T_STEPS = 8;

__global__ __launch_bounds__(256) void if_neuron_fwd(const v4f* __restrict__ x,
                                                     v4f* __restrict__ out,
                                                     int n4) {
    int i = blockIdx.x * blockDim.x + threadIdx.x;
    if (i >= n4) return;

    const size_t stride = (size_t)n4;   // v4f elements per timestep (uniform)

    // ---- Phase 1: issue all 8 independent B128 NT loads back-to-back ----
    v4f xv[T_STEPS];
#pragma unroll
    for (int t = 0; t < T_STEPS; ++t) {
        xv[t] = __builtin_nontemporal_load(x + (size_t)t * stride + (size_t)i);
    }
    // Hard scheduling fence: nothing may be reordered across this point, so
    // all 8 global_load_b128 are in flight before the first consumer. The
    // compiler then emits descending s_wait_loadcnt (7,6,...) at each use.
    __builtin_amdgcn_sched_barrier(0);

    // ---- Phase 2: sequential integrate/threshold/reset chain ----
    v4f mem = (v4f)0.0f;
#pragma unroll
    for (int t = 0; t < T_STEPS; ++t) {
        v4f o;
#pragma unroll
        for (int c = 0; c < 4; ++c) {
            float m = mem[c] + xv[t][c];      // integrate
            bool sp = (m > 1.0f);             // hard threshold (Vth = 1)
            o[c]    = sp ? 1.0f : 0.0f;       // spike output
            mem[c]  = sp ? 0.0f : m;          // hard reset
        }
        __builtin_nontemporal_store(o, out + (size_t)t * stride + (size_t)i);
    }
}

extern "C" void kernel_launch(void* const* d_in, const int* in_sizes, int n_in,
                              void* d_out, int out_size, void* d_ws, size_t ws_size,
                              hipStream_t stream) {
    const float* x = (const float*)d_in[0];
    float* out     = (float*)d_out;

    // x is [T=8, N] flat; N = 32*256*32*32 = 8,388,608 (divisible by 4).
    long long total = in_sizes[0];
    int n  = (int)(total / T_STEPS);   // elements per timestep
    int n4 = n / 4;                    // float4 columns per timestep

    const int block = 256;             // 8 wave32s per workgroup
    const int grid  = (n4 + block - 1) / block;

    if_neuron_fwd<<<grid, block, 0, stream>>>((const v4f*)x, (v4f*)out, n4);
}